// GCN_mamba_Net_57148834840956
// MI455X (gfx1250) — compile-verified
//
#include <hip/hip_runtime.h>
#include <hip/hip_bf16.h>

typedef __attribute__((ext_vector_type(2))) float v2f;
typedef __attribute__((ext_vector_type(8))) float v8f;

#define LSEQ   2048
#define FIN    512
#define DMODEL 256
#define DINNER 256
#define DTRANK 16
#define NSTATE 16
#define NLAYER 8
#define NCLS   40
#define HDIM_  8
#define NHEADS 32
#define DST2   8
#define DCONV  4
#define CONVD  272     // DINNER + 2*DST2
#define DPROJ  560     // 2*DINNER + 2*DST2 + NHEADS
#define XDBLW  48      // DTRANK + 2*NSTATE

__device__ __forceinline__ float sigmoid_(float x){ return 1.f/(1.f+__expf(-x)); }
__device__ __forceinline__ float silu_(float x){ return x*sigmoid_(x); }
__device__ __forceinline__ float softplus_(float x){ return (x>20.f)? x : log1pf(__expf(x)); }

#define WMMA_F32(acc,a,b) \
  acc = __builtin_amdgcn_wmma_f32_16x16x4_f32(false,(a),false,(b),(short)0,(acc),false,false)

// ------------------------------------------------------------------
// Generic fp32 WMMA GEMM: C[M,N] = act(alpha * A[M,K]@B[K,N] + beta*Cin)
// One wave computes a 16x64 tile via V_WMMA_F32_16X16X4_F32; block = 8 waves
// covering 128 rows. M multiple of 128, K multiple of 4; N edge-guarded.
// Fast path: full 64-wide tile, zero branches in the K loop.
// Edge path: clamped loads * {0,1} mask (cndmask, no exec manipulation).
// act: 0=none, 1=relu, 2=softplus.
// ------------------------------------------------------------------
__global__ __launch_bounds__(256)
void wmma_gemm_f32(const float* __restrict__ A, int lda,
                   const float* __restrict__ B, int ldb,
                   float* __restrict__ C, int ldc,
                   const float* __restrict__ Cin, int ldcin,
                   int M, int N, int K, float alpha, float beta, int act)
{
  const int lane = threadIdx.x & 31;
  const int wave = threadIdx.x >> 5;
  const int half = lane >> 4;     // selects K pair (A/B) and M half (C/D)
  const int l16  = lane & 15;
  const int row0 = blockIdx.x*128 + wave*16;
  const int col0 = blockIdx.y*64;
  if (row0 >= M) return;

  v8f acc0 = {0.f,0.f,0.f,0.f,0.f,0.f,0.f,0.f};
  v8f acc1 = acc0, acc2 = acc0, acc3 = acc0;

  // A fragment: lane l16 -> row m, half selects K pair {2h, 2h+1}
  const float* Ap = A + (size_t)(row0 + l16)*lda + 2*half;
  // B fragment base: row (k + 2*half), column col0 + l16
  const size_t ldbs = (size_t)ldb;
  const float* Bp = B + (size_t)(2*half)*ldbs + col0 + l16;

  if (col0 + 64 <= N) {
    // ---------------- full-tile fast path: no guards ----------------
    for (int k=0;k<K;k+=4){
      v2f a;  a.x = Ap[0];  a.y = Ap[1];
      v2f b0, b1, b2, b3;
      b0.x = Bp[0];   b0.y = Bp[ldbs];
      b1.x = Bp[16];  b1.y = Bp[ldbs+16];
      b2.x = Bp[32];  b2.y = Bp[ldbs+32];
      b3.x = Bp[48];  b3.y = Bp[ldbs+48];
      WMMA_F32(acc0, a, b0);
      WMMA_F32(acc1, a, b1);
      WMMA_F32(acc2, a, b2);
      WMMA_F32(acc3, a, b3);
      Ap += 4;  Bp += 4*ldbs;
    }
  } else {
    // ---------------- edge path: clamped load * mask -----------------
    int  coff[4];
    float cmask[4];
    #pragma unroll
    for (int j=0;j<4;++j){
      const int c = col0 + 16*j + l16;
      const bool val = (c < N);
      cmask[j] = val ? 1.f : 0.f;
      // Bp[coff[j]] == B[row*ldb + min(c, N-1)]  (always in-bounds)
      coff[j] = val ? 16*j : (N-1) - (col0 + l16);
    }
    for (int k=0;k<K;k+=4){
      v2f a;  a.x = Ap[0];  a.y = Ap[1];
      v2f b0, b1, b2, b3;
      b0.x = Bp[coff[0]]*cmask[0];  b0.y = Bp[ldbs+coff[0]]*cmask[0];
      b1.x = Bp[coff[1]]*cmask[1];  b1.y = Bp[ldbs+coff[1]]*cmask[1];
      b2.x = Bp[coff[2]]*cmask[2];  b2.y = Bp[ldbs+coff[2]]*cmask[2];
      b3.x = Bp[coff[3]]*cmask[3];  b3.y = Bp[ldbs+coff[3]]*cmask[3];
      WMMA_F32(acc0, a, b0);
      WMMA_F32(acc1, a, b1);
      WMMA_F32(acc2, a, b2);
      WMMA_F32(acc3, a, b3);
      Ap += 4;  Bp += 4*ldbs;
    }
  }

  // C/D layout: VGPR v -> row (row0 + v + 8*half), col (col0 + 16*j + l16)
  v8f accs[4] = {acc0, acc1, acc2, acc3};
  #pragma unroll
  for (int j=0;j<4;++j){
    const int c = col0 + 16*j + l16;
    if (c < N){
      #pragma unroll
      for (int v=0;v<8;++v){
        const int r = row0 + v + 8*half;
        float val = alpha*accs[j][v];
        if (Cin) val += beta*Cin[(size_t)r*ldcin + c];
        if (act==1)      val = fmaxf(val, 0.f);
        else if (act==2) val = softplus_(val);
        C[(size_t)r*ldc + c] = val;
      }
    }
  }
}

// ------------------- Mamba2 (bidirectional) helpers -------------------

// dt[t,h] = softplus(zx[t, 528+h] + dt_bias[h])
__global__ void m2_dt_kernel(const float* __restrict__ zx,
                             const float* __restrict__ dt_bias,
                             float* __restrict__ dt)
{
  int i = blockIdx.x*blockDim.x + threadIdx.x;
  if (i >= LSEQ*NHEADS) return;
  int h = i & (NHEADS-1);
  int t = i >> 5;
  dt[i] = softplus_(zx[(size_t)t*DPROJ + (DPROJ-NHEADS) + h] + dt_bias[h]);
}

// depthwise causal conv (k=4) + SiLU, direction handled by index mapping
__global__ void m2_conv_kernel(const float* __restrict__ zx,
                               const float* __restrict__ w,
                               const float* __restrict__ b,
                               float* __restrict__ out, int dir)
{
  int i = blockIdx.x*blockDim.x + threadIdx.x;
  if (i >= LSEQ*CONVD) return;
  int c = i % CONVD;
  int t = i / CONVD;
  float acc = b[c];
  #pragma unroll
  for (int k=0;k<DCONV;++k){
    int src = dir ? (t + (DCONV-1) - k) : (t - (DCONV-1) + k);
    if (src >= 0 && src < LSEQ)
      acc += zx[(size_t)src*DPROJ + DINNER + c] * w[c*DCONV + k];
  }
  out[i] = silu_(acc);
}

// sequential SSM scan; blockIdx.x = direction, thread = (head,hdim) pair
__global__ __launch_bounds__(256)
void m2_scan_kernel(const float* __restrict__ xbc_f,
                    const float* __restrict__ xbc_r,
                    const float* __restrict__ dt,
                    const float* __restrict__ A_log,
                    const float* __restrict__ Dp,
                    float* __restrict__ y_f, float* __restrict__ y_r)
{
  const int dir = blockIdx.x;
  const float* xbc = dir ? xbc_r : xbc_f;
  float* y = dir ? y_r : y_f;
  const int tid = threadIdx.x;        // h*8 + p
  const int h = tid >> 3;
  const float Ah = -__expf(A_log[h]);
  const float Dh = Dp[h];
  float s[DST2];
  #pragma unroll
  for (int n=0;n<DST2;++n) s[n]=0.f;

  for (int step=0; step<LSEQ; ++step){
    const int t = dir ? (LSEQ-1-step) : step;
    const float* row = xbc + (size_t)t*CONVD;
    const float xv  = row[tid];
    const float dtv = dt[t*NHEADS + h];
    const float dAv = __expf(dtv * Ah);
    const float coef = dtv * xv;
    float acc = Dh * xv;
    #pragma unroll
    for (int n=0;n<DST2;++n){
      s[n] = dAv*s[n] + coef*row[DINNER + n];          // B
      acc += s[n]*row[DINNER + DST2 + n];              // C
    }
    y[(size_t)t*DINNER + tid] = acc;
  }
}

// g = (y * silu(z)) * rsqrt(mean(g^2)+eps) * norm_w   (in-place over y)
__global__ __launch_bounds__(256)
void m2_gatenorm_kernel(const float* __restrict__ zx,
                        const float* __restrict__ y_f, const float* __restrict__ y_r,
                        const float* __restrict__ norm_w,
                        float* __restrict__ g_f, float* __restrict__ g_r)
{
  const int t = blockIdx.x;
  const int dir = blockIdx.y;
  const float* y = dir ? y_r : y_f;
  float* g = dir ? g_r : g_f;
  const int d = threadIdx.x;
  const float zv = zx[(size_t)t*DPROJ + d];
  const float gv = y[(size_t)t*DINNER + d] * silu_(zv);
  __shared__ float red[256];
  red[d] = gv*gv;
  __syncthreads();
  for (int off=128; off>0; off>>=1){
    if (d<off) red[d] += red[d+off];
    __syncthreads();
  }
  const float inv = rsqrtf(red[0]*(1.f/DINNER) + 1e-5f);
  g[(size_t)t*DINNER + d] = gv*inv*norm_w[d];
}

// g_out = relu(m2sum)*0.9 + 0.1*x_input ; xs[:,0,:] = relu(bn1(x_input))
__global__ void post_m2_kernel(const float* __restrict__ m2sum,
                               const float* __restrict__ x_input,
                               const float* __restrict__ bn1_g,
                               const float* __restrict__ bn1_b,
                               float* __restrict__ g_out,
                               float* __restrict__ xs)
{
  int i = blockIdx.x*blockDim.x + threadIdx.x;
  if (i >= LSEQ*DMODEL) return;
  const int d = i & (DMODEL-1);
  const int t = i >> 8;
  const float xin = x_input[i];
  g_out[i] = fmaxf(m2sum[i], 0.f)*0.9f + 0.1f*xin;
  const float bscale = bn1_g[d] * rsqrtf(1.f + 1e-5f);
  const float xb = fmaxf(xin*bscale + bn1_b[d], 0.f);
  xs[(size_t)t*(NLAYER*DINNER) + d] = xb;     // layer 0 slice
}

// ------------------- GCN-Mamba selective scan (8 steps) -------------------
__global__ __launch_bounds__(256)
void blk_scan_kernel(const float* __restrict__ xs,
                     const float* __restrict__ xdbl,
                     const float* __restrict__ delta,
                     const float* __restrict__ A_log,
                     const float* __restrict__ Dp,
                     float* __restrict__ yout)
{
  int i = blockIdx.x*blockDim.x + threadIdx.x;   // i = b*256 + d
  if (i >= LSEQ*DINNER) return;
  const int d = i & (DINNER-1);
  const int b = i >> 8;
  float An[NSTATE];
  #pragma unroll
  for (int n=0;n<NSTATE;++n) An[n] = -__expf(A_log[d*NSTATE + n]);
  const float Dd = Dp[d];
  float s[NSTATE];
  #pragma unroll
  for (int n=0;n<NSTATE;++n) s[n]=0.f;

  const float* xsb = xs    + (size_t)b*NLAYER*DINNER;
  const float* xdb = xdbl  + (size_t)b*NLAYER*XDBLW;
  const float* deb = delta + (size_t)b*NLAYER*DINNER;
  float* yb        = yout  + (size_t)b*NLAYER*DINNER;

  for (int l=0;l<NLAYER;++l){
    const float dv = deb[l*DINNER + d];
    const float xv = xsb[l*DINNER + d];
    const float coef = dv*xv;
    float acc = Dd*xv;
    const float* xrow = xdb + l*XDBLW;
    #pragma unroll
    for (int n=0;n<NSTATE;++n){
      const float dA = __expf(dv*An[n]);
      s[n] = dA*s[n] + coef*xrow[DTRANK + n];            // B
      acc += s[n]*xrow[DTRANK + NSTATE + n];             // C
    }
    yb[l*DINNER + d] = acc;
  }
}

// output = relu(bn2((all_raw[:,7,:] + x_input)*0.5 + g_out*0.5))
__global__ void final_out_kernel(const float* __restrict__ all_raw,
                                 const float* __restrict__ x_input,
                                 const float* __restrict__ g_out,
                                 const float* __restrict__ bn2_g,
                                 const float* __restrict__ bn2_b,
                                 float* __restrict__ outvec)
{
  int i = blockIdx.x*blockDim.x + threadIdx.x;
  if (i >= LSEQ*DMODEL) return;
  const int d = i & (DMODEL-1);
  const int t = i >> 8;
  float v = (all_raw[(size_t)t*(NLAYER*DINNER) + (NLAYER-1)*DINNER + d] + x_input[i])*0.5f
            + g_out[i]*0.5f;
  v = v * (bn2_g[d]*rsqrtf(1.f + 1e-5f)) + bn2_b[d];
  outvec[i] = fmaxf(v, 0.f);
}

__global__ void relu_inplace_kernel(float* __restrict__ p, int n)
{
  int i = blockIdx.x*blockDim.x + threadIdx.x;
  if (i < n) p[i] = fmaxf(p[i], 0.f);
}

__global__ __launch_bounds__(64)
void logsoftmax_kernel(const float* __restrict__ logits, float* __restrict__ out)
{
  const int t = blockIdx.x;
  const int k = threadIdx.x;
  __shared__ float red[64];
  const float v = (k < NCLS) ? logits[(size_t)t*NCLS + k] : -1e30f;
  red[k] = v; __syncthreads();
  for (int off=32; off>0; off>>=1){ if (k<off) red[k]=fmaxf(red[k],red[k+off]); __syncthreads(); }
  const float mx = red[0]; __syncthreads();
  red[k] = (k < NCLS) ? __expf(v - mx) : 0.f; __syncthreads();
  for (int off=32; off>0; off>>=1){ if (k<off) red[k]+=red[k+off]; __syncthreads(); }
  const float lse = __logf(red[0]);
  if (k < NCLS) out[(size_t)t*NCLS + k] = v - mx - lse;
}

// ----------------------------------------------------------------------
extern "C" void kernel_launch(void* const* d_in, const int* in_sizes, int n_in,
                              void* d_out, int out_size, void* d_ws, size_t ws_size,
                              hipStream_t stream)
{
  const float* x         = (const float*)d_in[0];
  const float* adj       = (const float*)d_in[1];
  const float* lin1_W    = (const float*)d_in[2];
  const float* in_proj   = (const float*)d_in[3];
  const float* conv_w    = (const float*)d_in[4];
  const float* conv_b    = (const float*)d_in[5];
  const float* dt_bias   = (const float*)d_in[6];
  const float* m2_A_log  = (const float*)d_in[7];
  const float* m2_D      = (const float*)d_in[8];
  const float* norm_w    = (const float*)d_in[9];
  const float* out_proj  = (const float*)d_in[10];
  const float* xproj_W   = (const float*)d_in[11];
  const float* dtproj_W  = (const float*)d_in[12];
  const float* blk_A_log = (const float*)d_in[13];
  const float* blk_D     = (const float*)d_in[14];
  const float* outproj_W = (const float*)d_in[15];
  const float* bn1_g     = (const float*)d_in[16];
  const float* bn1_b     = (const float*)d_in[17];
  const float* bn2_g     = (const float*)d_in[18];
  const float* bn2_b     = (const float*)d_in[19];
  const float* lin2_W    = (const float*)d_in[20];
  (void)in_sizes; (void)n_in; (void)out_size; (void)ws_size;

  float* out = (float*)d_out;                 // [all_out(2048*8*256) | logsm(2048*40)]

  float* ws = (float*)d_ws;
  size_t off = 0;
  float* x_input = ws + off; off += (size_t)LSEQ*DMODEL;
  float* zx      = ws + off; off += (size_t)LSEQ*DPROJ;
  float* dtbuf   = ws + off; off += (size_t)LSEQ*NHEADS;
  float* xbcf    = ws + off; off += (size_t)LSEQ*CONVD;
  float* xbcr    = ws + off; off += (size_t)LSEQ*CONVD;
  float* yf      = ws + off; off += (size_t)LSEQ*DINNER;
  float* yr      = ws + off; off += (size_t)LSEQ*DINNER;
  float* m2sum   = ws + off; off += (size_t)LSEQ*DMODEL;
  float* g_out   = ws + off; off += (size_t)LSEQ*DMODEL;
  float* xs      = ws + off; off += (size_t)LSEQ*NLAYER*DINNER;
  float* xdbl    = ws + off; off += (size_t)LSEQ*NLAYER*XDBLW;
  float* delta   = ws + off; off += (size_t)LSEQ*NLAYER*DINNER;
  float* yblk    = ws + off; off += (size_t)LSEQ*NLAYER*DINNER;
  float* outvec  = ws + off; off += (size_t)LSEQ*DMODEL;
  float* logits  = ws + off; off += (size_t)LSEQ*NCLS;

  auto gemm = [&](const float* A,int lda,const float* B,int ldb,
                  float* C,int ldc,const float* Cin,int ldcin,
                  int M,int N,int K,float al,float be,int act){
    dim3 g(M/128, (N+63)/64);
    wmma_gemm_f32<<<g, 256, 0, stream>>>(A,lda,B,ldb,C,ldc,Cin,ldcin,M,N,K,al,be,act);
  };

  // 1) x_input = x @ lin1_W
  gemm(x, FIN, lin1_W, DMODEL, x_input, DMODEL, nullptr, 0, LSEQ, DMODEL, FIN, 1.f, 0.f, 0);
  // 2) zxbcdt = x_input @ in_proj
  gemm(x_input, DMODEL, in_proj, DPROJ, zx, DPROJ, nullptr, 0, LSEQ, DPROJ, DMODEL, 1.f, 0.f, 0);
  // 3) dt = softplus(raw + bias)
  m2_dt_kernel<<<(LSEQ*NHEADS+255)/256, 256, 0, stream>>>(zx, dt_bias, dtbuf);
  // 4) conv + silu, both directions
  m2_conv_kernel<<<(LSEQ*CONVD+255)/256, 256, 0, stream>>>(zx, conv_w, conv_b, xbcf, 0);
  m2_conv_kernel<<<(LSEQ*CONVD+255)/256, 256, 0, stream>>>(zx, conv_w, conv_b, xbcr, 1);
  // 5) SSM scans (fwd + rev), results at original positions
  m2_scan_kernel<<<2, 256, 0, stream>>>(xbcf, xbcr, dtbuf, m2_A_log, m2_D, yf, yr);
  // 6) gate * RMS-norm (in place over yf/yr)
  m2_gatenorm_kernel<<<dim3(LSEQ,2), 256, 0, stream>>>(zx, yf, yr, norm_w, yf, yr);
  // 7) m2sum = gf@out_proj + gr@out_proj
  gemm(yf, DINNER, out_proj, DMODEL, m2sum, DMODEL, nullptr, 0, LSEQ, DMODEL, DINNER, 1.f, 0.f, 0);
  gemm(yr, DINNER, out_proj, DMODEL, m2sum, DMODEL, m2sum, DMODEL, LSEQ, DMODEL, DINNER, 1.f, 1.f, 0);
  // 8) g_out + xb (= xs layer 0)
  post_m2_kernel<<<(LSEQ*DMODEL+255)/256, 256, 0, stream>>>(m2sum, x_input, bn1_g, bn1_b, g_out, xs);
  // 9) GCN propagation: xs[:,i,:] = 0.95*(adj @ xs[:,i-1,:]) + 0.05*xb
  for (int i=1;i<NLAYER;++i)
    gemm(adj, LSEQ, xs + (size_t)(i-1)*DINNER, NLAYER*DINNER,
         xs + (size_t)i*DINNER, NLAYER*DINNER,
         xs, NLAYER*DINNER, LSEQ, DINNER, LSEQ, 0.95f, 0.05f, 0);
  // 10) x_dbl = relu(xs @ xproj_W)
  gemm(xs, DINNER, xproj_W, XDBLW, xdbl, XDBLW, nullptr, 0, LSEQ*NLAYER, XDBLW, DINNER, 1.f, 0.f, 1);
  // 11) delta = softplus(x_dbl[:, :16] @ dtproj_W)
  gemm(xdbl, XDBLW, dtproj_W, DINNER, delta, DINNER, nullptr, 0, LSEQ*NLAYER, DINNER, DTRANK, 1.f, 0.f, 2);
  // 12) selective scan over 8 layers
  blk_scan_kernel<<<(LSEQ*DINNER+255)/256, 256, 0, stream>>>(xs, xdbl, delta, blk_A_log, blk_D, yblk);
  // 13) all_raw = yblk @ outproj_W  (written straight into d_out, pre-relu)
  gemm(yblk, DINNER, outproj_W, DMODEL, out, DMODEL, nullptr, 0, LSEQ*NLAYER, DMODEL, DINNER, 1.f, 0.f, 0);
  // 14) output vector (uses pre-relu last layer of all_raw)
  final_out_kernel<<<(LSEQ*DMODEL+255)/256, 256, 0, stream>>>(out, x_input, g_out, bn2_g, bn2_b, outvec);
  // 15) logits = output @ lin2_W
  gemm(outvec, DMODEL, lin2_W, NCLS, logits, NCLS, nullptr, 0, LSEQ, NCLS, DMODEL, 1.f, 0.f, 0);
  // 16) all_out = relu(all_raw) in place
  relu_inplace_kernel<<<(LSEQ*NLAYER*DINNER+255)/256, 256, 0, stream>>>(out, LSEQ*NLAYER*DINNER);
  // 17) log_softmax into d_out tail
  logsoftmax_kernel<<<LSEQ, 64, 0, stream>>>(logits, out + (size_t)LSEQ*NLAYER*DINNER);
}